// NetworkLSTM_2817498546219
// MI455X (gfx1250) — compile-verified
//
#include <hip/hip_runtime.h>
#include <math.h>

// Problem sizes (fixed by the reference)
#define IN_SIZE  4096
#define HIDDEN   4096
#define CATK     8192   // IN_SIZE + HIDDEN
#define OUT_SIZE 4096

typedef __attribute__((ext_vector_type(2))) float v2f;
typedef __attribute__((ext_vector_type(8))) float v8f;

// 16 waves per block: K split 16 ways per 16-row tile for load concurrency.
#define WPB 16
#define TPB (WPB * 32)

// ---------------------------------------------------------------------------
// Bandwidth-bound GEMV tile via V_WMMA_F32_16X16X4_F32 (full fp32 precision).
//
// One wave owns rows [rowtile*16, rowtile*16+16) and a K-slice of length K/WPB.
// A-fragment (16x4 f32, 2 VGPRs), per ISA layout:
//   lanes 0-15 : M = lane,     vgpr0 = K0, vgpr1 = K1
//   lanes 16-31: M = lane-16,  vgpr0 = K2, vgpr1 = K3
// -> each lane does one global b64 load of W[row, k+koff .. k+koff+1].
// B-fragment: xh[k..k+3] replicated across all 16 N columns
//   (vgpr0 = xh[k] / xh[k+2] per lane half, vgpr1 = xh[k+1] / xh[k+3]),
// so D[m, n] is the same valid partial dot product for every n; we harvest
// lanes 0 and 16 (rows vgpr / 8+vgpr) and reduce the WPB K-slices in LDS.
//
// The wave index is forced scalar via readfirstlane so the K-loop is a pure
// SALU loop: EXEC stays all-1s across every WMMA (ISA requirement) and the
// hot loop carries no per-iteration VALU compare / exec masking.
// ---------------------------------------------------------------------------
template <int K>
__device__ __forceinline__ void gemv16_wmma_tile(const float* __restrict__ W,
                                                 const float* __restrict__ x,
                                                 const float* __restrict__ bias,
                                                 float* __restrict__ y,
                                                 int rowtile) {
    const int lane    = threadIdx.x & 31;
    const int wave    = __builtin_amdgcn_readfirstlane(threadIdx.x) >> 5; // wave-uniform scalar
    const int rowbase = rowtile * 16;
    const int row     = rowbase + (lane & 15);
    const int koff    = (lane >> 4) << 1;          // 0 (lanes 0-15) or 2 (lanes 16-31)

    constexpr int KC = K / WPB;                    // K-slice per wave (compile-time)
    const int     k0 = wave * KC;

    const float* wrow = W + (size_t)row * (size_t)K + (size_t)(k0 + koff);
    const float* xk   = x + k0 + koff;

    v8f acc = {};
#pragma unroll 16
    for (int k = 0; k < KC; k += 4) {
        v2f a = *(const v2f*)(wrow + k);           // 256B of W per wave per WMMA
        v2f b = *(const v2f*)(xk + k);             // broadcast xh chunk (L0/L2 hit)
        acc = __builtin_amdgcn_wmma_f32_16x16x4_f32(
            /*neg_a=*/false, a, /*neg_b=*/false, b,
            /*c_mod=*/(short)0, acc, /*reuse_a=*/false, /*reuse_b=*/false);
    }

    __shared__ float red[WPB * 16];
    if (lane == 0 || lane == 16) {
        const int mbase = (lane >> 4) * 8;         // lane0 -> rows 0-7, lane16 -> 8-15
#pragma unroll
        for (int m = 0; m < 8; ++m)
            red[wave * 16 + mbase + m] = acc[m];
    }
    __syncthreads();

    if (threadIdx.x < 16) {
        float s = bias[rowbase + threadIdx.x];
#pragma unroll
        for (int w = 0; w < WPB; ++w)
            s += red[w * 16 + threadIdx.x];
        y[rowbase + threadIdx.x] = s;
    }
}

// Fused 4-gate GEMV: g[gate*HIDDEN + r] = W_gate[r,:] @ xh + b_gate[r]
// grid = (HIDDEN/16, 4); all four weight matrices stream concurrently.
__global__ void __launch_bounds__(TPB)
lstm_gates_gemv(const float* __restrict__ Wf, const float* __restrict__ bf,
                const float* __restrict__ Wi, const float* __restrict__ bi,
                const float* __restrict__ Wa, const float* __restrict__ ba,
                const float* __restrict__ Wo, const float* __restrict__ bo,
                const float* __restrict__ xh, float* __restrict__ g) {
    const float* W;
    const float* b;
    switch (blockIdx.y) {
        default:
        case 0: W = Wf; b = bf; break;
        case 1: W = Wi; b = bi; break;
        case 2: W = Wa; b = ba; break;
        case 3: W = Wo; b = bo; break;
    }
    gemv16_wmma_tile<CATK>(W, xh, b, g + (size_t)blockIdx.y * HIDDEN, blockIdx.x);
}

// out = Wout @ new_hidden + bout ; grid = (OUT_SIZE/16, 1)
__global__ void __launch_bounds__(TPB)
lstm_out_gemv(const float* __restrict__ Wout, const float* __restrict__ bout,
              const float* __restrict__ nh, float* __restrict__ out) {
    gemv16_wmma_tile<HIDDEN>(Wout, nh, bout, out, blockIdx.x);
}

// xh = concat(x, hidden)  (8192 elements)
__global__ void lstm_concat(const float* __restrict__ x,
                            const float* __restrict__ h,
                            float* __restrict__ xh) {
    int j = blockIdx.x * blockDim.x + threadIdx.x;
    if (j < CATK)
        xh[j] = (j < IN_SIZE) ? x[j] : h[j - IN_SIZE];
}

// Pointwise LSTM cell update (4096 elements):
//   forget = sig(f); update = sig(i)*tanh(a); nc = forget*cell + update;
//   nh = tanh(nc) * sig(o)
__global__ void lstm_pointwise(const float* __restrict__ g,
                               const float* __restrict__ cell,
                               float* __restrict__ nh) {
    int j = blockIdx.x * blockDim.x + threadIdx.x;
    if (j >= HIDDEN) return;
    float f = g[j];
    float i = g[HIDDEN + j];
    float a = g[2 * HIDDEN + j];
    float o = g[3 * HIDDEN + j];
    float fs = 1.0f / (1.0f + expf(-f));
    float is = 1.0f / (1.0f + expf(-i));
    float os = 1.0f / (1.0f + expf(-o));
    float at = tanhf(a);
    float nc = fs * cell[j] + is * at;
    nh[j] = tanhf(nc) * os;
}

extern "C" void kernel_launch(void* const* d_in, const int* in_sizes, int n_in,
                              void* d_out, int out_size, void* d_ws, size_t ws_size,
                              hipStream_t stream) {
    (void)in_sizes; (void)n_in; (void)out_size; (void)ws_size;

    const float* x      = (const float*)d_in[0];
    const float* hidden = (const float*)d_in[1];
    const float* cell   = (const float*)d_in[2];
    const float* Wf     = (const float*)d_in[3];
    const float* bf     = (const float*)d_in[4];
    const float* Wi     = (const float*)d_in[5];
    const float* bi     = (const float*)d_in[6];
    const float* Wa     = (const float*)d_in[7];
    const float* ba     = (const float*)d_in[8];
    const float* Wo     = (const float*)d_in[9];
    const float* bo     = (const float*)d_in[10];
    const float* Wout   = (const float*)d_in[11];
    const float* bout   = (const float*)d_in[12];
    float*       out    = (float*)d_out;

    // Workspace layout (floats): [0,8192) xh | [8192,24576) gates g | [24576,28672) nh
    float* ws = (float*)d_ws;
    float* xh = ws;
    float* g  = ws + CATK;
    float* nh = ws + CATK + 4 * HIDDEN;

    lstm_concat<<<CATK / 256, 256, 0, stream>>>(x, hidden, xh);

    lstm_gates_gemv<<<dim3(HIDDEN / 16, 4), TPB, 0, stream>>>(
        Wf, bf, Wi, bi, Wa, ba, Wo, bo, xh, g);

    lstm_pointwise<<<HIDDEN / 256, 256, 0, stream>>>(g, cell, nh);

    lstm_out_gemv<<<OUT_SIZE / 16, TPB, 0, stream>>>(Wout, bout, nh, out);
}